// SetRelationshipLoss_47614007443964
// MI455X (gfx1250) — compile-verified
//
#include <hip/hip_runtime.h>
#include <hip/hip_bf16.h>

// ---------------------------------------------------------------------------
// SetRelationshipLoss for MI455X (gfx1250, wave32)
//   B=8, N=2048, D=256, E=4096
// Pipeline:
//   1. row_norms           : sq[n] = ||x_n||^2
//   2. gram_pass<false>    : WMMA f32 Gram tiles -> d2 -> 16-bit-prefix histogram
//   3. scan_hist           : find threshold bin T, below-count, k_rem
//   4. gram_pass<true>     : recompute tiles, collect bin<T pairs + bin==T candidates
//   5. select_tail         : bitonic-sort candidates (bits,idx), take k_rem smallest
//   6. edge_keys           : key = (x[src]-x[dst]) . w  (one wave per edge)
//   7. sort_edges          : per-(tensor,batch) bitonic sort of 4096 (key,pair)
//   8. loss_partial        : sum ((r-t)^2) per block (deterministic partials)
//   9. final_reduce        : deterministic tree sum -> mean -> d_out[0]
// GEMM: 128x64 block tile, 8 waves, each wave a 2x2 grid of 16x16 WMMA tiles
// (A/B fragment reuse -> 4 v_wmma per 4 LDS fragment loads).
// ---------------------------------------------------------------------------

#define BB    8
#define NN    2048
#define DD    256
#define EE    4096
#define NTB   16            // 2 tensors * 8 batches
#define NBIN  65536
#define CAND_CAP 4096

#define TI    128           // block tile rows
#define TJ    64            // block tile cols
#define KC    64            // K chunk staged in LDS
#define LDSP  68            // padded LDS row stride (floats)

typedef __attribute__((ext_vector_type(2))) float v2f;
typedef __attribute__((ext_vector_type(8))) float v8f;

// ---------------- workspace layout (bytes) ----------------
#define SQ_OFF    ((size_t)0)                       // 2*8*2048*4      = 131072
#define HIST_OFF  ((size_t)131072)                  // 16*65536*4      = 4194304
#define CNT_OFF   ((size_t)(HIST_OFF + 4194304))    // 16*2*4          = 128
#define TH_OFF    ((size_t)(CNT_OFF + 128))         // 16*4*4          = 256
#define SEL_OFF   ((size_t)(TH_OFF + 256))          // 16*4096*4       = 262144
#define CAND_OFF  ((size_t)(SEL_OFF + 262144))      // 16*4096*8       = 524288
#define KEY_OFF   ((size_t)(CAND_OFF + 524288))     // 16*4096*8       = 524288
#define ORD_OFF   ((size_t)(KEY_OFF + 524288))      // 16*4096*4       = 262144
#define PART_OFF  ((size_t)(ORD_OFF + 262144))      // 4096*4          = 16384

// ---------------------------------------------------------------------------
__device__ __forceinline__ float wave_sum32(float v) {
  #pragma unroll
  for (int off = 16; off > 0; off >>= 1) v += __shfl_xor(v, off, 32);
  return v;
}

__device__ __forceinline__ void bitonic_u64(unsigned long long* s, int n) {
  for (int k = 2; k <= n; k <<= 1) {
    for (int j = k >> 1; j > 0; j >>= 1) {
      for (int i = (int)threadIdx.x; i < n; i += (int)blockDim.x) {
        int l = i ^ j;
        if (l > i) {
          bool up = ((i & k) == 0);
          unsigned long long a = s[i], b = s[l];
          if ((a > b) == up) { s[i] = b; s[l] = a; }
        }
      }
      __syncthreads();
    }
  }
}

// ---------------------------------------------------------------------------
// 1. row norms: one wave per row, rows = 2*8*2048
__global__ __launch_bounds__(256) void srl_row_norms(
    const float* __restrict__ recon, const float* __restrict__ target,
    float* __restrict__ sqout) {
  int gid  = blockIdx.x * 8 + (threadIdx.x >> 5);   // 0..32767
  int lane = threadIdx.x & 31;
  int t    = gid >> 14;                             // 16384 rows per tensor
  int rem  = gid & 16383;                           // b*N + n
  const float* row = (t ? target : recon) + (size_t)rem * DD;
  float s = 0.f;
  for (int d = lane; d < DD; d += 32) { float v = row[d]; s = __builtin_fmaf(v, v, s); }
  s = wave_sum32(s);
  if (lane == 0) sqout[gid] = s;
}

// ---------------------------------------------------------------------------
// 2/4. fused WMMA Gram + d2 epilogue.  COLLECT=false: histogram; true: collect.
template <bool COLLECT>
__global__ __launch_bounds__(256) void srl_gram_pass(
    const float* __restrict__ recon, const float* __restrict__ target,
    const float* __restrict__ sq_all,
    unsigned* __restrict__ hist_all,
    const unsigned* __restrict__ thresh,
    unsigned* __restrict__ cnt,
    unsigned* __restrict__ sel,
    unsigned long long* __restrict__ cand) {
  __shared__ float As[TI * LDSP];   // 128 x 68 floats
  __shared__ float Bs[TJ * LDSP];   //  64 x 68 floats

  const int tb = blockIdx.z;                 // 0..15
  const int t  = tb >> 3, b = tb & 7;
  const float* __restrict__ x = (t ? target : recon) + (size_t)b * NN * DD;
  const float* __restrict__ sq = sq_all + (size_t)tb * NN;

  const int i0   = blockIdx.y * TI;
  const int j0   = blockIdx.x * TJ;
  const int tid  = threadIdx.x;
  const int lane = tid & 31, wave = tid >> 5;
  const int wr   = wave & 3, wc = wave >> 2; // wave tile: rows wr*32, cols wc*32
  const int half = lane >> 4, lm = lane & 15;

  v8f acc[2][2];
  #pragma unroll
  for (int ti = 0; ti < 2; ++ti)
    #pragma unroll
    for (int tj = 0; tj < 2; ++tj)
      acc[ti][tj] = (v8f){0.f, 0.f, 0.f, 0.f, 0.f, 0.f, 0.f, 0.f};

  for (int k0 = 0; k0 < DD; k0 += KC) {
    // stage A (128 x 64) and B (64 x 64) tiles, float4-coalesced
    #pragma unroll
    for (int q = 0; q < 8; ++q) {
      int idx = tid + q * 256;               // 0..2047
      int r = idx >> 4, c4 = (idx & 15) << 2;
      const float* gp = x + (size_t)(i0 + r) * DD + k0 + c4;
      if (q == 0 && k0 + KC < DD)
        __builtin_prefetch(gp + KC, 0, 0);   // global_prefetch_b8 next chunk
      float4 v = *reinterpret_cast<const float4*>(gp);
      float* dst = &As[r * LDSP + c4];
      dst[0] = v.x; dst[1] = v.y; dst[2] = v.z; dst[3] = v.w;
    }
    #pragma unroll
    for (int q = 0; q < 4; ++q) {
      int idx = tid + q * 256;               // 0..1023
      int r = idx >> 4, c4 = (idx & 15) << 2;
      const float* gp = x + (size_t)(j0 + r) * DD + k0 + c4;
      if (q == 0 && k0 + KC < DD)
        __builtin_prefetch(gp + KC, 0, 0);
      float4 v = *reinterpret_cast<const float4*>(gp);
      float* dst = &Bs[r * LDSP + c4];
      dst[0] = v.x; dst[1] = v.y; dst[2] = v.z; dst[3] = v.w;
    }
    __syncthreads();

    // 16 WMMA steps of K=4 over this chunk; 4 wmma per step (2x2 tile reuse)
    #pragma unroll
    for (int k = 0; k < KC; k += 4) {
      const int ka = k + half * 2;           // ISA A-layout: lanes0-15 K=k,k+1; lanes16-31 K=k+2,k+3
      v2f a0, a1, b0, b1;
      a0[0] = As[(wr * 32 + lm)      * LDSP + ka];
      a0[1] = As[(wr * 32 + lm)      * LDSP + ka + 1];
      a1[0] = As[(wr * 32 + 16 + lm) * LDSP + ka];
      a1[1] = As[(wr * 32 + 16 + lm) * LDSP + ka + 1];
      b0[0] = Bs[(wc * 32 + lm)      * LDSP + ka];
      b0[1] = Bs[(wc * 32 + lm)      * LDSP + ka + 1];
      b1[0] = Bs[(wc * 32 + 16 + lm) * LDSP + ka];
      b1[1] = Bs[(wc * 32 + 16 + lm) * LDSP + ka + 1];
#if __has_builtin(__builtin_amdgcn_wmma_f32_16x16x4_f32)
      acc[0][0] = __builtin_amdgcn_wmma_f32_16x16x4_f32(false, a0, false, b0, (short)0, acc[0][0], false, false);
      acc[0][1] = __builtin_amdgcn_wmma_f32_16x16x4_f32(false, a0, false, b1, (short)0, acc[0][1], false, false);
      acc[1][0] = __builtin_amdgcn_wmma_f32_16x16x4_f32(false, a1, false, b0, (short)0, acc[1][0], false, false);
      acc[1][1] = __builtin_amdgcn_wmma_f32_16x16x4_f32(false, a1, false, b1, (short)0, acc[1][1], false, false);
#else
      acc[0][0][0] = __builtin_fmaf(a0[0], b0[0], acc[0][0][0]);
      acc[0][1][0] = __builtin_fmaf(a0[0], b1[0], acc[0][1][0]);
      acc[1][0][0] = __builtin_fmaf(a1[0], b0[0], acc[1][0][0]);
      acc[1][1][0] = __builtin_fmaf(a1[0], b1[0], acc[1][1][0]);
#endif
    }
    __syncthreads();
  }

  // epilogue: C layout -> lane holds col lm, rows v + 8*half (per 16x16 tile)
  #pragma unroll
  for (int tj = 0; tj < 2; ++tj) {
    const int jcol  = j0 + wc * 32 + tj * 16 + lm;
    const float sqj = sq[jcol];
    #pragma unroll
    for (int ti = 0; ti < 2; ++ti) {
      const int i_base = i0 + wr * 32 + ti * 16 + half * 8;
      if (!COLLECT) {
        unsigned* __restrict__ hist = hist_all + (size_t)tb * NBIN;
        #pragma unroll
        for (int v = 0; v < 8; ++v) {
          int i = i_base + v;
          float d2 = sq[i] + sqj - 2.0f * acc[ti][tj][v];
          if (i == jcol) d2 = __builtin_inff();
          unsigned bin = __float_as_uint(d2) >> 16;   // d2 >= 0 -> bits monotone
          atomicAdd(&hist[bin], 1u);
        }
      } else {
        const unsigned T = thresh[tb * 4 + 0];
        #pragma unroll
        for (int v = 0; v < 8; ++v) {
          int i = i_base + v;
          float d2 = sq[i] + sqj - 2.0f * acc[ti][tj][v];
          if (i == jcol) d2 = __builtin_inff();
          unsigned bits = __float_as_uint(d2);
          unsigned bin  = bits >> 16;
          unsigned pk   = (unsigned)(i * NN + jcol);  // flattened idx (src*N + dst)
          if (bin < T) {
            unsigned pos = atomicAdd(&cnt[tb * 2 + 0], 1u);
            sel[(size_t)tb * EE + pos] = pk;
          } else if (bin == T) {
            unsigned pos = atomicAdd(&cnt[tb * 2 + 1], 1u);
            if (pos < CAND_CAP)
              cand[(size_t)tb * CAND_CAP + pos] =
                  ((unsigned long long)bits << 32) | (unsigned long long)pk;
          }
        }
      }
    }
  }
}

// ---------------------------------------------------------------------------
// 3. scan histogram: find threshold bin T with cum >= 4096
__global__ __launch_bounds__(1024) void srl_scan_hist(
    const unsigned* __restrict__ hist_all, unsigned* __restrict__ thresh) {
  __shared__ unsigned sc[1024];
  const int tb = blockIdx.x;
  const unsigned* __restrict__ hist = hist_all + (size_t)tb * NBIN;
  const int base = (int)threadIdx.x * 64;
  unsigned local = 0;
  for (int i = 0; i < 64; ++i) local += hist[base + i];
  sc[threadIdx.x] = local;
  __syncthreads();
  // Hillis-Steele inclusive scan over 1024 chunk sums
  for (int off = 1; off < 1024; off <<= 1) {
    unsigned v   = sc[threadIdx.x];
    unsigned add = (threadIdx.x >= (unsigned)off) ? sc[threadIdx.x - off] : 0u;
    __syncthreads();
    sc[threadIdx.x] = v + add;
    __syncthreads();
  }
  const unsigned K = EE;
  unsigned incl = sc[threadIdx.x];
  unsigned excl = incl - local;
  if (excl < K && incl >= K) {
    unsigned cum = excl;
    for (int i = 0; i < 64; ++i) {
      unsigned c = hist[base + i];
      if (cum + c >= K) {
        thresh[tb * 4 + 0] = (unsigned)(base + i); // T
        thresh[tb * 4 + 1] = K - cum;              // k_rem from bin T
        thresh[tb * 4 + 2] = cum;                  // count strictly below
        thresh[tb * 4 + 3] = 0u;
        break;
      }
      cum += c;
    }
  }
}

// ---------------------------------------------------------------------------
// 5. sort threshold-bin candidates by (bits, idx), append k_rem smallest
__global__ __launch_bounds__(256) void srl_select_tail(
    const unsigned long long* __restrict__ cand,
    const unsigned* __restrict__ cnt,
    const unsigned* __restrict__ thresh,
    unsigned* __restrict__ sel) {
  __shared__ unsigned long long s[CAND_CAP];     // 32 KB
  const int tb = blockIdx.x;
  const unsigned krem  = thresh[tb * 4 + 1];
  const unsigned below = thresh[tb * 4 + 2];
  unsigned cc = cnt[tb * 2 + 1];
  if (cc > CAND_CAP) cc = CAND_CAP;
  for (int i = (int)threadIdx.x; i < CAND_CAP; i += 256)
    s[i] = (i < (int)cc) ? cand[(size_t)tb * CAND_CAP + i] : ~0ULL;
  __syncthreads();
  bitonic_u64(s, CAND_CAP);
  for (int i = (int)threadIdx.x; i < (int)krem; i += 256)
    sel[(size_t)tb * EE + below + i] = (unsigned)(s[i] & 0xFFFFFFFFu);
}

// ---------------------------------------------------------------------------
// 6. projection keys: one wave per selected edge
__global__ __launch_bounds__(256) void srl_edge_keys(
    const float* __restrict__ recon, const float* __restrict__ target,
    const float* __restrict__ w, const unsigned* __restrict__ sel,
    unsigned long long* __restrict__ keys) {
  int gid  = blockIdx.x * 8 + (threadIdx.x >> 5); // 0..65535
  int lane = threadIdx.x & 31;
  int tb   = gid >> 12;
  int e    = gid & (EE - 1);
  int t    = tb >> 3, b = tb & 7;
  const float* __restrict__ x = (t ? target : recon) + (size_t)b * NN * DD;
  unsigned p = sel[(size_t)tb * EE + e];
  int src = (int)(p / NN), dst = (int)(p % NN);
  float s = 0.f;
  for (int d = lane; d < DD; d += 32)
    s = __builtin_fmaf(x[(size_t)src * DD + d] - x[(size_t)dst * DD + d], w[d], s);
  s = wave_sum32(s);
  if (lane == 0) {
    unsigned kb = __float_as_uint(s);
    kb = (kb & 0x80000000u) ? ~kb : (kb | 0x80000000u);  // monotone for signed floats
    keys[(size_t)tb * EE + e] = ((unsigned long long)kb << 32) | (unsigned long long)p;
  }
}

// ---------------------------------------------------------------------------
// 7. canonical ordering: bitonic sort 4096 (key,pair) per (tensor,batch)
__global__ __launch_bounds__(256) void srl_sort_edges(
    const unsigned long long* __restrict__ keys, unsigned* __restrict__ ord) {
  __shared__ unsigned long long s[EE];            // 32 KB
  const int tb = blockIdx.x;
  for (int i = (int)threadIdx.x; i < EE; i += 256) s[i] = keys[(size_t)tb * EE + i];
  __syncthreads();
  bitonic_u64(s, EE);
  for (int i = (int)threadIdx.x; i < EE; i += 256)
    ord[(size_t)tb * EE + i] = (unsigned)(s[i] & 0xFFFFFFFFu);
}

// ---------------------------------------------------------------------------
// 8. loss partials (deterministic; no float atomics)
__global__ __launch_bounds__(256) void srl_loss_partial(
    const float* __restrict__ recon, const float* __restrict__ target,
    const unsigned* __restrict__ ord, float* __restrict__ part) {
  int gid  = blockIdx.x * 8 + (threadIdx.x >> 5); // 0..32767 : b*4096 + e
  int lane = threadIdx.x & 31;
  int b = gid >> 12, e = gid & (EE - 1);
  const float* __restrict__ xr = recon  + (size_t)b * NN * DD;
  const float* __restrict__ xt = target + (size_t)b * NN * DD;
  unsigned rp = ord[(size_t)(0 * BB + b) * EE + e];
  unsigned tp = ord[(size_t)(1 * BB + b) * EE + e];
  int rs = (int)(rp / NN), rd = (int)(rp % NN);
  int ts = (int)(tp / NN), td = (int)(tp % NN);
  float s = 0.f;
  for (int d = lane; d < DD; d += 32) {
    float diff = (xr[(size_t)rs * DD + d] - xr[(size_t)rd * DD + d])
               - (xt[(size_t)ts * DD + d] - xt[(size_t)td * DD + d]);
    s = __builtin_fmaf(diff, diff, s);
  }
  s = wave_sum32(s);
  __shared__ float wsum[8];
  if (lane == 0) wsum[threadIdx.x >> 5] = s;
  __syncthreads();
  if (threadIdx.x == 0) {
    float accv = 0.f;
    #pragma unroll
    for (int i = 0; i < 8; ++i) accv += wsum[i];
    part[blockIdx.x] = accv;
  }
}

// 9. deterministic final reduction -> mean
__global__ __launch_bounds__(256) void srl_final_reduce(
    const float* __restrict__ part, float* __restrict__ out) {
  __shared__ double red[256];
  double s = 0.0;
  for (int i = (int)threadIdx.x; i < 4096; i += 256) s += (double)part[i];
  red[threadIdx.x] = s;
  __syncthreads();
  for (int off = 128; off > 0; off >>= 1) {
    if ((int)threadIdx.x < off) red[threadIdx.x] += red[threadIdx.x + off];
    __syncthreads();
  }
  if (threadIdx.x == 0)
    out[0] = (float)(red[0] / ((double)BB * EE * DD));
}

// ---------------------------------------------------------------------------
extern "C" void kernel_launch(void* const* d_in, const int* in_sizes, int n_in,
                              void* d_out, int out_size, void* d_ws, size_t ws_size,
                              hipStream_t stream) {
  (void)in_sizes; (void)n_in; (void)out_size; (void)ws_size;
  const float* recon  = (const float*)d_in[0];
  const float* target = (const float*)d_in[1];
  const float* w      = (const float*)d_in[2];

  char* ws = (char*)d_ws;
  float*              sq   = (float*)(ws + SQ_OFF);
  unsigned*           hist = (unsigned*)(ws + HIST_OFF);
  unsigned*           cnt  = (unsigned*)(ws + CNT_OFF);
  unsigned*           th   = (unsigned*)(ws + TH_OFF);
  unsigned*           sel  = (unsigned*)(ws + SEL_OFF);
  unsigned long long* cand = (unsigned long long*)(ws + CAND_OFF);
  unsigned long long* keys = (unsigned long long*)(ws + KEY_OFF);
  unsigned*           ord  = (unsigned*)(ws + ORD_OFF);
  float*              part = (float*)(ws + PART_OFF);

  // zero histograms + counters (contiguous region)
  hipMemsetAsync(ws + HIST_OFF, 0, (size_t)NTB * NBIN * 4 + (size_t)NTB * 2 * 4, stream);

  srl_row_norms<<<4096, 256, 0, stream>>>(recon, target, sq);

  dim3 gg(NN / TJ, NN / TI, NTB);   // 32 x 16 x 16
  srl_gram_pass<false><<<gg, 256, 0, stream>>>(recon, target, sq, hist, th, cnt, sel, cand);
  srl_scan_hist<<<NTB, 1024, 0, stream>>>(hist, th);
  srl_gram_pass<true><<<gg, 256, 0, stream>>>(recon, target, sq, hist, th, cnt, sel, cand);
  srl_select_tail<<<NTB, 256, 0, stream>>>(cand, cnt, th, sel);
  srl_edge_keys<<<(2 * BB * EE) / 8, 256, 0, stream>>>(recon, target, w, sel, keys);
  srl_sort_edges<<<NTB, 256, 0, stream>>>(keys, ord);
  srl_loss_partial<<<(BB * EE) / 8, 256, 0, stream>>>(recon, target, ord, part);
  srl_final_reduce<<<1, 256, 0, stream>>>(part, (float*)d_out);
}